// TrafficRuleNet_63909113364904
// MI455X (gfx1250) — compile-verified
//
#include <hip/hip_runtime.h>
#include <math.h>

typedef float v2f __attribute__((ext_vector_type(2)));
typedef float v8f __attribute__((ext_vector_type(8)));

// ---------------------------------------------------------------------------
// 1) init: deg = 1.0 (self loop), pooled = 0, cnt = 0
// ---------------------------------------------------------------------------
__global__ void init_kernel(float* __restrict__ deg, float* __restrict__ pooled,
                            float* __restrict__ cnt, int n, int gch, int g) {
    int i = blockIdx.x * blockDim.x + threadIdx.x;
    if (i < n)   deg[i]    = 1.0f;
    if (i < gch) pooled[i] = 0.0f;
    if (i < g)   cnt[i]    = 0.0f;
}

// ---------------------------------------------------------------------------
// 2) in-degree accumulation over edges
// ---------------------------------------------------------------------------
__global__ void degree_kernel(const int* __restrict__ ei, float* __restrict__ deg, int e) {
    int i = blockIdx.x * blockDim.x + threadIdx.x;
    if (i >= e) return;
    int d = ei[e + i];            // edge_index[1][i]
    atomicAdd(deg + d, 1.0f);
}

// ---------------------------------------------------------------------------
// 3) dinv = rsqrt(deg)  (in place)
// ---------------------------------------------------------------------------
__global__ void rsqrt_kernel(float* __restrict__ deg, int n) {
    int i = blockIdx.x * blockDim.x + threadIdx.x;
    if (i >= n) return;
    deg[i] = rsqrtf(deg[i]);
}

// ---------------------------------------------------------------------------
// 4) GEMM1 via V_WMMA_F32_16X16X4_F32:  t1 = x[N,3] @ W1[3,16]  (K padded 3->4)
//    also writes agg1 = dinv^2 * t1 (self-loop term). One wave per 16-node tile.
//    Branchless operand build; wave-uniform full-tile store fast path.
// ---------------------------------------------------------------------------
__global__ void gemm1_wmma(const float* __restrict__ x, const float* __restrict__ W1,
                           const float* __restrict__ dinv,
                           float* __restrict__ t1, float* __restrict__ agg1,
                           int n, int ntiles) {
    int wave = (blockIdx.x * blockDim.x + threadIdx.x) >> 5;
    int lane = threadIdx.x & 31;
    if (wave >= ntiles) return;                 // wave-uniform exit, EXEC stays full
    int base = wave * 16;
    int half = lane >> 4;                       // 0: K=0,1   1: K=2,3(pad)
    int l16  = lane & 15;
    int row  = base + l16; if (row > n - 1) row = n - 1;

    // A 16x4 layout: lo lanes hold K=0,1; hi lanes hold K=2,3. K=3 padded to 0.
    float xa = x[row * 3 + 2 * half];           // K=0 (lo) or K=2 (hi)
    float xb = x[row * 3 + 1];                  // K=1 (lo only)
    v2f a; a.x = xa; a.y = half ? 0.0f : xb;
    // B 4x16 layout mirrors A: lo lanes rows 0,1; hi lanes rows 2,3(pad).
    float wa = W1[(2 * half) * 16 + l16];
    float wb = W1[1 * 16 + l16];
    v2f b; b.x = wa; b.y = half ? 0.0f : wb;

    v8f c = {};
    c = __builtin_amdgcn_wmma_f32_16x16x4_f32(false, a, false, b, (short)0, c, false, false);

    int mbase = base + half * 8;                // D layout: vgpr r -> M = r (+8 for hi lanes)
    if (base + 16 <= n) {                       // wave-uniform: full tile, no guards
#pragma unroll
        for (int r = 0; r < 8; ++r) {
            int m = mbase + r;
            float v  = c[r];
            float di = dinv[m];
            t1[(size_t)m * 16 + l16]   = v;
            agg1[(size_t)m * 16 + l16] = di * di * v;
        }
    } else {
#pragma unroll
        for (int r = 0; r < 8; ++r) {
            int m = mbase + r;
            if (m < n) {
                float v  = c[r];
                float di = dinv[m];
                t1[(size_t)m * 16 + l16]   = v;
                agg1[(size_t)m * 16 + l16] = di * di * v;
            }
        }
    }
}

// ---------------------------------------------------------------------------
// 5/8) edge scatter: agg[dst] += dinv[src]*dinv[dst] * t[src]  (C channels)
// ---------------------------------------------------------------------------
template <int C>
__global__ void edge_scatter(const int* __restrict__ ei, const float* __restrict__ dinv,
                             const float* __restrict__ t, float* __restrict__ agg, int e) {
    int i = blockIdx.x * blockDim.x + threadIdx.x;
    if (i >= e) return;
    int s = ei[i];
    int d = ei[e + i];
    float norm = dinv[s] * dinv[d];
    const float4* ts = reinterpret_cast<const float4*>(t + (size_t)s * C);
    float* ad = agg + (size_t)d * C;
#pragma unroll
    for (int q = 0; q < C / 4; ++q) {
        float4 v = ts[q];
        atomicAdd(ad + q * 4 + 0, norm * v.x);
        atomicAdd(ad + q * 4 + 1, norm * v.y);
        atomicAdd(ad + q * 4 + 2, norm * v.z);
        atomicAdd(ad + q * 4 + 3, norm * v.w);
    }
}

// ---------------------------------------------------------------------------
// 6) h = relu(agg + bias)   (in place, C power of two)
// ---------------------------------------------------------------------------
__global__ void bias_relu(float* __restrict__ h, const float* __restrict__ b,
                          int total, int cmask) {
    int i = blockIdx.x * blockDim.x + threadIdx.x;
    if (i >= total) return;
    float v = h[i] + b[i & cmask];
    h[i] = v > 0.0f ? v : 0.0f;
}

// ---------------------------------------------------------------------------
// 7) GEMM2 via WMMA: t2 = h1[N,16] @ W2[16,32]; agg2 = dinv^2 * t2
//    One wave per 16-node tile; K=16 -> 4 chained WMMAs; N=32 -> 2 acc chains.
// ---------------------------------------------------------------------------
__global__ void gemm2_wmma(const float* __restrict__ h1, const float* __restrict__ W2,
                           const float* __restrict__ dinv,
                           float* __restrict__ t2, float* __restrict__ agg2,
                           int n, int ntiles) {
    int wave = (blockIdx.x * blockDim.x + threadIdx.x) >> 5;
    int lane = threadIdx.x & 31;
    if (wave >= ntiles) return;
    int base = wave * 16;
    int half = lane >> 4;
    int l16  = lane & 15;
    int row  = base + l16; if (row > n - 1) row = n - 1;
    const float* hr = h1 + (size_t)row * 16;

    v8f acc0 = {}, acc1 = {};
#pragma unroll
    for (int kc = 0; kc < 4; ++kc) {
        int k0 = kc * 4 + half * 2;             // A layout: lo lanes K=0,1; hi lanes K=2,3
        v2f a;  a.x = hr[k0];                  a.y = hr[k0 + 1];
        v2f b0; b0.x = W2[k0 * 32 + l16];      b0.y = W2[(k0 + 1) * 32 + l16];
        v2f b1; b1.x = W2[k0 * 32 + 16 + l16]; b1.y = W2[(k0 + 1) * 32 + 16 + l16];
        acc0 = __builtin_amdgcn_wmma_f32_16x16x4_f32(false, a, false, b0, (short)0, acc0, false, false);
        acc1 = __builtin_amdgcn_wmma_f32_16x16x4_f32(false, a, false, b1, (short)0, acc1, false, false);
    }

    int mbase = base + half * 8;
    if (base + 16 <= n) {                       // wave-uniform: full tile, no guards
#pragma unroll
        for (int r = 0; r < 8; ++r) {
            int m = mbase + r;
            float di = dinv[m];
            float d2 = di * di;
            float v0 = acc0[r], v1 = acc1[r];
            t2[(size_t)m * 32 + l16]        = v0;
            t2[(size_t)m * 32 + 16 + l16]   = v1;
            agg2[(size_t)m * 32 + l16]      = d2 * v0;
            agg2[(size_t)m * 32 + 16 + l16] = d2 * v1;
        }
    } else {
#pragma unroll
        for (int r = 0; r < 8; ++r) {
            int m = mbase + r;
            if (m < n) {
                float di = dinv[m];
                float d2 = di * di;
                float v0 = acc0[r], v1 = acc1[r];
                t2[(size_t)m * 32 + l16]        = v0;
                t2[(size_t)m * 32 + 16 + l16]   = v1;
                agg2[(size_t)m * 32 + l16]      = d2 * v0;
                agg2[(size_t)m * 32 + 16 + l16] = d2 * v1;
            }
        }
    }
}

// ---------------------------------------------------------------------------
// 9) relu(agg2 + b2) + global mean-pool accumulate.
//    batch is sorted -> two-stage: LDS (ds_add_f32) window of 8 graphs anchored
//    at the block's first graph, global-atomic fallback, one flush per block.
// ---------------------------------------------------------------------------
#define POOL_WIN 8
__global__ void relu_pool(const float* __restrict__ agg2, const float* __restrict__ b2,
                          const int* __restrict__ batch, float* __restrict__ pooled,
                          float* __restrict__ cnt, int n) {
    __shared__ float lpool[POOL_WIN * 32];
    __shared__ float lcnt[POOL_WIN];
    int t  = threadIdx.x;
    int i0 = blockIdx.x * blockDim.x;
    if (t < POOL_WIN) lcnt[t] = 0.0f;
    for (int j = t; j < POOL_WIN * 32; j += blockDim.x) lpool[j] = 0.0f;
    int gbase = batch[i0 < n ? i0 : (n - 1)];   // window anchor (sorted batch)
    __syncthreads();

    int i = i0 + t;
    if (i < n) {
        int g   = batch[i];
        int rel = g - gbase;
        const float* row = agg2 + (size_t)i * 32;
        if (rel >= 0 && rel < POOL_WIN) {
            atomicAdd(&lcnt[rel], 1.0f);
            float* lp = &lpool[rel * 32];
#pragma unroll
            for (int c = 0; c < 32; ++c) {
                float v = row[c] + b2[c];
                v = v > 0.0f ? v : 0.0f;
                atomicAdd(lp + c, v);
            }
        } else {                                // rare straggler path
            atomicAdd(cnt + g, 1.0f);
            float* pg = pooled + (size_t)g * 32;
#pragma unroll
            for (int c = 0; c < 32; ++c) {
                float v = row[c] + b2[c];
                v = v > 0.0f ? v : 0.0f;
                atomicAdd(pg + c, v);
            }
        }
    }
    __syncthreads();

    if (t < POOL_WIN && lcnt[t] > 0.0f) atomicAdd(cnt + gbase + t, lcnt[t]);
    for (int j = t; j < POOL_WIN * 32; j += blockDim.x) {
        float v = lpool[j];
        if (v != 0.0f) atomicAdd(pooled + (size_t)gbase * 32 + j, v);
    }
}

// ---------------------------------------------------------------------------
// 10) per-graph MLP head + log_softmax  (512 graphs; tiny)
// ---------------------------------------------------------------------------
__global__ void head_kernel(const float* __restrict__ pooled, const float* __restrict__ cnt,
                            const float* __restrict__ fW1, const float* __restrict__ fb1,
                            const float* __restrict__ fW2, const float* __restrict__ fb2,
                            float* __restrict__ out, int g) {
    int i = blockIdx.x * blockDim.x + threadIdx.x;
    if (i >= g) return;
    float inv = 1.0f / fmaxf(cnt[i], 1.0f);
    float p[32];
#pragma unroll
    for (int c = 0; c < 32; ++c) p[c] = pooled[(size_t)i * 32 + c] * inv;
    float z[64];
#pragma unroll 4
    for (int j = 0; j < 64; ++j) {
        float s = fb1[j];
#pragma unroll
        for (int c = 0; c < 32; ++c) s += p[c] * fW1[c * 64 + j];
        z[j] = s > 0.0f ? s : 0.0f;
    }
    float o[3];
#pragma unroll
    for (int k = 0; k < 3; ++k) {
        float s = fb2[k];
#pragma unroll 4
        for (int j = 0; j < 64; ++j) s += z[j] * fW2[j * 3 + k];
        o[k] = s;
    }
    float m = fmaxf(o[0], fmaxf(o[1], o[2]));
    float l = logf(expf(o[0] - m) + expf(o[1] - m) + expf(o[2] - m));
#pragma unroll
    for (int k = 0; k < 3; ++k) out[i * 3 + k] = o[k] - m - l;
}

// ---------------------------------------------------------------------------
extern "C" void kernel_launch(void* const* d_in, const int* in_sizes, int n_in,
                              void* d_out, int out_size, void* d_ws, size_t ws_size,
                              hipStream_t stream) {
    const float* x     = (const float*)d_in[0];
    const int*   ei    = (const int*)  d_in[1];
    const int*   batch = (const int*)  d_in[2];
    const float* W1    = (const float*)d_in[3];
    const float* b1    = (const float*)d_in[4];
    const float* W2    = (const float*)d_in[5];
    const float* b2    = (const float*)d_in[6];
    const float* fW1   = (const float*)d_in[7];
    const float* fb1   = (const float*)d_in[8];
    const float* fW2   = (const float*)d_in[9];
    const float* fb2   = (const float*)d_in[10];
    float* out = (float*)d_out;

    const int n = in_sizes[0] / 3;      // 500000
    const int e = in_sizes[1] / 2;      // 8000000
    const int g = out_size / 3;         // 512

    // workspace carve-out (256B aligned regions)
    char* ws = (char*)d_ws;
    size_t off = 0;
    auto carve = [&](size_t bytes) -> float* {
        float* p = (float*)(ws + off);
        off = (off + bytes + 255) & ~(size_t)255;
        return p;
    };
    float* dinv   = carve((size_t)n * 4);            // deg -> dinv in place
    float* t1     = carve((size_t)n * 16 * 4);
    float* agg1   = carve((size_t)n * 16 * 4);       // becomes h1 after bias+relu
    float* t2     = carve((size_t)n * 32 * 4);
    float* agg2   = carve((size_t)n * 32 * 4);
    float* pooled = carve((size_t)g * 32 * 4);
    float* cnt    = carve((size_t)g * 4);
    (void)ws_size;

    const int B = 256;
    int ntiles = (n + 15) / 16;

    // 1) init
    init_kernel<<<(n + B - 1) / B, B, 0, stream>>>(dinv, pooled, cnt, n, g * 32, g);
    // 2) degree
    degree_kernel<<<(e + B - 1) / B, B, 0, stream>>>(ei, dinv, e);
    // 3) rsqrt
    rsqrt_kernel<<<(n + B - 1) / B, B, 0, stream>>>(dinv, n);
    // 4) layer-1 feature transform (WMMA f32) + self-loop term
    gemm1_wmma<<<(ntiles + 7) / 8, B, 0, stream>>>(x, W1, dinv, t1, agg1, n, ntiles);
    // 5) layer-1 edge scatter
    edge_scatter<16><<<(e + B - 1) / B, B, 0, stream>>>(ei, dinv, t1, agg1, e);
    // 6) bias + relu -> h1
    bias_relu<<<((n * 16) + B - 1) / B, B, 0, stream>>>(agg1, b1, n * 16, 15);
    // 7) layer-2 feature transform (WMMA f32) + self-loop term
    gemm2_wmma<<<(ntiles + 7) / 8, B, 0, stream>>>(agg1, W2, dinv, t2, agg2, n, ntiles);
    // 8) layer-2 edge scatter
    edge_scatter<32><<<(e + B - 1) / B, B, 0, stream>>>(ei, dinv, t2, agg2, e);
    // 9) bias + relu + pooled accumulate (LDS two-stage)
    relu_pool<<<(n + B - 1) / B, B, 0, stream>>>(agg2, b2, batch, pooled, cnt, n);
    // 10) MLP head + log_softmax
    head_kernel<<<(g + B - 1) / B, B, 0, stream>>>(pooled, cnt, fW1, fb1, fW2, fb2, out, g);
}